// GCN_pre_43654047596701
// MI455X (gfx1250) — compile-verified
//
#include <hip/hip_runtime.h>
#include <hip/hip_bf16.h>
#include <stdint.h>

typedef float v2f __attribute__((ext_vector_type(2)));
typedef float v8f __attribute__((ext_vector_type(8)));

#define DF 128  // feature dim for both layers (D_FEAT == NHID == 128)

// ---------------- graph normalization ----------------

__global__ __launch_bounds__(256) void k_deg_init(float* deg, int N) {
    int i = blockIdx.x * 256 + threadIdx.x;
    if (i < N) deg[i] = 1.0f;  // self-loop contributes 1 to every node
}

__global__ __launch_bounds__(256) void k_deg_edges(const long long* __restrict__ dst,
                                                   float* deg, int E) {
    int e = blockIdx.x * 256 + threadIdx.x;
    if (e < E) atomicAdd(&deg[(int)dst[e]], 1.0f);
}

__global__ __launch_bounds__(256) void k_dis(const float* __restrict__ deg,
                                             float* dis, int N) {
    int i = blockIdx.x * 256 + threadIdx.x;
    if (i < N) {
        float d = deg[i];
        dis[i] = (d > 0.0f) ? rsqrtf(d) : 0.0f;
    }
}

__global__ __launch_bounds__(256) void k_norm(const long long* __restrict__ src,
                                              const long long* __restrict__ dst,
                                              const float* __restrict__ dis,
                                              float* nrm, int E) {
    int e = blockIdx.x * 256 + threadIdx.x;
    if (e < E) nrm[e] = dis[(int)src[e]] * dis[(int)dst[e]];
}

// ---------------- fp32 WMMA GEMM: C[M x 128] = A[M x 128] * W[128 x 128] ----------------
// One block per 16-row tile; 8 waves, wave w -> 16-column tile w.
// A tile staged in LDS with padded stride (132) to avoid bank conflicts.

#define SA_STRIDE 132

__global__ __launch_bounds__(256) void k_gemm_wmma(const float* __restrict__ A,
                                                   const float* __restrict__ W,
                                                   float* __restrict__ C, int M) {
    __shared__ float sA[16 * SA_STRIDE];

    const int row0 = blockIdx.x * 16;
    const int lane = threadIdx.x & 31;
    const int wave = threadIdx.x >> 5;   // 0..7 -> column tile
    const int col0 = wave * 16;

    // cooperative load of the 16x128 A tile (shared by all 8 waves)
    for (int i = threadIdx.x; i < 16 * DF; i += 256) {
        int r = i >> 7;        // 0..15
        int c = i & (DF - 1);  // 0..127
        int gr = row0 + r;
        sA[r * SA_STRIDE + c] = (gr < M) ? A[(size_t)gr * DF + c] : 0.0f;
    }
    __syncthreads();

    // ISA VGPR layouts (wave32):
    //  A 16x4 fp32:  lane m in [0,15] / [16,31] -> M = lane&15 ; K = 2*(lane>>4) + {0,1}
    //  B 4x16 fp32:  lane   -> N = lane&15 ; K = 2*(lane>>4) + {0,1}
    //  C/D 16x16:    reg r  -> M = r + 8*(lane>>4) ; N = lane&15
    const int mrow    = lane & 15;
    const int halfsel = lane >> 4;

    v8f acc = {};
#pragma unroll
    for (int k0 = 0; k0 < DF; k0 += 4) {
        const int ka = k0 + 2 * halfsel;
        v2f a, b;
        a.x = sA[mrow * SA_STRIDE + ka];
        a.y = sA[mrow * SA_STRIDE + ka + 1];
        b.x = W[(size_t)ka * DF + col0 + mrow];
        b.y = W[(size_t)(ka + 1) * DF + col0 + mrow];
        acc = __builtin_amdgcn_wmma_f32_16x16x4_f32(
            /*neg_a=*/false, a, /*neg_b=*/false, b,
            /*c_mod=*/(short)0, acc, /*reuse_a=*/false, /*reuse_b=*/false);
    }

#pragma unroll
    for (int r = 0; r < 8; ++r) {
        int gr = row0 + r + 8 * halfsel;
        if (gr < M) C[(size_t)gr * DF + col0 + mrow] = acc[r];
    }
}

// ---------------- aggregation ----------------

// out[i,f] = X[i,f] * dis[i]^2   (self-loop term; also initializes the buffer)
__global__ __launch_bounds__(256) void k_self_init(const float* __restrict__ X,
                                                   const float* __restrict__ dis,
                                                   float* out, int N) {
    int idx = blockIdx.x * 256 + threadIdx.x;
    int i = idx >> 7;
    int f = idx & (DF - 1);
    if (i < N) {
        float di = dis[i];
        out[(size_t)i * DF + f] = X[(size_t)i * DF + f] * di * di;
    }
}

// out[dst,f] += X[src,f] * norm[e]   (128 threads per edge; L2-resident fp32 atomics)
__global__ __launch_bounds__(256) void k_scatter(const long long* __restrict__ src,
                                                 const long long* __restrict__ dst,
                                                 const float* __restrict__ nrm,
                                                 const float* __restrict__ X,
                                                 float* out, int E) {
    int idx = blockIdx.x * 256 + threadIdx.x;
    int e = idx >> 7;
    int f = idx & (DF - 1);
    if (e < E) {
        int s = (int)src[e];
        int d = (int)dst[e];
        float w = nrm[e];
        atomicAdd(&out[(size_t)d * DF + f], X[(size_t)s * DF + f] * w);
    }
}

__global__ __launch_bounds__(256) void k_bias_relu(const float* __restrict__ X,
                                                   const float* __restrict__ b,
                                                   float* out, int total) {
    int idx = blockIdx.x * 256 + threadIdx.x;
    if (idx < total) {
        float v = X[idx] + b[idx & (DF - 1)];
        out[idx] = v > 0.0f ? v : 0.0f;
    }
}

__global__ __launch_bounds__(256) void k_bias_out(const float* __restrict__ X,
                                                  const float* __restrict__ b,
                                                  float* out, int total) {
    int idx = blockIdx.x * 256 + threadIdx.x;
    if (idx < total) out[idx] = X[idx] + b[idx & (DF - 1)];
}

// ---------------- driver ----------------

extern "C" void kernel_launch(void* const* d_in, const int* in_sizes, int n_in,
                              void* d_out, int out_size, void* d_ws, size_t ws_size,
                              hipStream_t stream) {
    const float*     x   = (const float*)d_in[0];
    const long long* ei  = (const long long*)d_in[1];  // int64 edge_index [2, E]
    const float*     w1  = (const float*)d_in[2];
    const float*     b1  = (const float*)d_in[3];
    const float*     w2  = (const float*)d_in[4];
    const float*     b2  = (const float*)d_in[5];

    const int N = in_sizes[0] / DF;
    const int E = in_sizes[1] / 2;
    const long long* src = ei;
    const long long* dst = ei + E;

    // workspace layout (floats): deg[N] | dis[N] | nrm[E] | bufA[N*DF] | bufB[N*DF]
    float* ws   = (float*)d_ws;
    float* deg  = ws;
    float* dis  = deg + N;
    float* nrm  = dis + N;
    float* bufA = nrm + E;
    float* bufB = bufA + (size_t)N * DF;

    const int nbN   = (N + 255) / 256;
    const int nbE   = (E + 255) / 256;
    const int nbNF  = (int)(((long long)N * DF + 255) / 256);
    const int nbEF  = (int)(((long long)E * DF + 255) / 256);
    const int tiles = (N + 15) / 16;

    // graph normalization (once, shared by both layers)
    k_deg_init <<<nbN, 256, 0, stream>>>(deg, N);
    k_deg_edges<<<nbE, 256, 0, stream>>>(dst, deg, E);
    k_dis      <<<nbN, 256, 0, stream>>>(deg, dis, N);
    k_norm     <<<nbE, 256, 0, stream>>>(src, dst, dis, nrm, E);

    // layer 1: xw = x@w1 ; agg = scatter(norm * gather(xw)) ; h = relu(agg + b1)
    k_gemm_wmma<<<tiles, 256, 0, stream>>>(x, w1, bufA, N);
    k_self_init<<<nbNF, 256, 0, stream>>>(bufA, dis, bufB, N);
    k_scatter  <<<nbEF, 256, 0, stream>>>(src, dst, nrm, bufA, bufB, E);
    k_bias_relu<<<nbNF, 256, 0, stream>>>(bufB, b1, bufA, N * DF);

    // layer 2: hw = h@w2 ; agg2 = scatter(...) ; out = agg2 + b2
    k_gemm_wmma<<<tiles, 256, 0, stream>>>(bufA, w2, bufB, N);
    k_self_init<<<nbNF, 256, 0, stream>>>(bufB, dis, bufA, N);
    k_scatter  <<<nbEF, 256, 0, stream>>>(src, dst, nrm, bufB, bufA, E);
    k_bias_out <<<nbNF, 256, 0, stream>>>(bufA, b2, (float*)d_out, N * DF);
}